// GCN_68049461838507
// MI455X (gfx1250) — compile-verified
//
#include <hip/hip_runtime.h>
#include <hip/hip_bf16.h>
#include <math.h>

typedef __attribute__((ext_vector_type(2))) float v2f;
typedef __attribute__((ext_vector_type(8))) float v8f;
typedef __attribute__((ext_vector_type(4))) unsigned int u32x4;
typedef __attribute__((ext_vector_type(8))) int i32x8;
typedef __attribute__((ext_vector_type(4))) int i32x4;

// ---------------------------------------------------------------- utilities
__global__ void gcn_zero_f32(float* __restrict__ p, int n) {
    int i = blockIdx.x * blockDim.x + threadIdx.x;
    if (i < n) p[i] = 0.0f;
}

// deg[col[e]] += 1  (atomic, L2-resident 200KB array)
__global__ void gcn_degree(const long long* __restrict__ col,
                           float* __restrict__ deg, int E) {
    int e = blockIdx.x * blockDim.x + threadIdx.x;
    if (e < E) {
        int c = (int)col[e];
        __hip_atomic_fetch_add(&deg[c], 1.0f, __ATOMIC_RELAXED,
                               __HIP_MEMORY_SCOPE_AGENT);
    }
}

// dinv[i] = rsqrt(deg[i] + 1)   (+1 = self loop; always > 0)
__global__ void gcn_dinv(float* __restrict__ deg, int N) {
    int i = blockIdx.x * blockDim.x + threadIdx.x;
    if (i < N) deg[i] = rsqrtf(deg[i] + 1.0f);
}

// ---------------------------------------------------------------- WMMA GEMM
// H[N,128] = act(A[N,128]) @ W[128,128]
// One block = one 16-row M tile. A-tile (8KB) staged by the Tensor Data Mover
// with pad-on-load (4 DWORDs every 128 DWORDs) -> LDS layout [16][132],
// bank-conflict-free for the column reads. 8 waves, wave w owns N-columns
// [16w,16w+16). K loop: 32 x V_WMMA_F32_16X16X4_F32, fully unrolled.
template <bool RELU>
__global__ __launch_bounds__(256)
void gcn_gemm_wmma(const float* __restrict__ A, const float* __restrict__ W,
                   float* __restrict__ H, int N) {
    __shared__ float Ash[16][132];

    const int m0   = blockIdx.x * 16;
    const int wave = threadIdx.x >> 5;
    const int lane = threadIdx.x & 31;
    const int n0   = wave * 16;

    if (wave == 0) {
        // ---- Tensor DMA descriptor (D#) for the 16x128 f32 tile ----
        unsigned long long gaddr =
            (unsigned long long)(const void*)(A + (size_t)m0 * 128);
        unsigned int ldsa = (unsigned int)(size_t)(&Ash[0][0]);
        unsigned int tdim1 = (unsigned int)(N - m0);   // rows left (OOB->0)

        u32x4 g0;
        g0[0] = 1u;                                   // count=1, user desc
        g0[1] = ldsa;                                 // lds_addr
        g0[2] = (unsigned int)(gaddr & 0xFFFFFFFFu);  // global_addr[31:0]
        g0[3] = (unsigned int)((gaddr >> 32) & 0x01FFFFFFu) | (2u << 30); // +type=2

        i32x8 g1 = {};
        // data_size=2 (4B), pad_enable=1, pad_interval=6 (128 DW), pad_amount=3 (4 DW)
        g1[0] = (int)((2u << 16) | (1u << 20) | (6u << 22) | (3u << 25));
        g1[1] = (int)(128u << 16);                    // tensor_dim0 = 128 (bits 79:48)
        g1[2] = (int)((tdim1 & 0xFFFFu) << 16);       // tensor_dim1 lo (bits 111:80)
        g1[3] = (int)((128u << 16) | ((tdim1 >> 16) & 0xFFFFu)); // tile_dim0=128 | tdim1 hi
        g1[4] = 16;                                   // tile_dim1 = 16 rows
        g1[5] = 128;                                  // tensor_dim0_stride = 128
        g1[6] = 0;
        g1[7] = 0;

        i32x4 g2 = {};   // 2D tensor: groups 2/3 unused
        i32x4 g3 = {};
        i32x8 g4 = {};   // trailing group (clang-23 6-arg form), unused

        __builtin_amdgcn_tensor_load_to_lds(g0, g1, g2, g3, g4, /*cpol=*/0);
        __builtin_amdgcn_s_wait_tensorcnt(0);
    }
    __syncthreads();   // publish TDM-written LDS to all 8 waves

    const int r     = lane & 15;           // M index (A) / N index (B,C)
    const int khalf = (lane < 16) ? 0 : 2; // K sub-offset for this half-wave
    const float* Wn = W + n0 + r;          // column strip of W

    v8f acc = {};
#pragma unroll
    for (int kk = 0; kk < 128; kk += 4) {
        v2f a, b;
        float a0 = Ash[r][kk + khalf];
        float a1 = Ash[r][kk + khalf + 1];
        if (RELU) { a0 = fmaxf(a0, 0.0f); a1 = fmaxf(a1, 0.0f); }
        a.x = a0;
        a.y = a1;
        b.x = Wn[(size_t)(kk + khalf) * 128];
        b.y = Wn[(size_t)(kk + khalf + 1) * 128];
        acc = __builtin_amdgcn_wmma_f32_16x16x4_f32(
            /*neg_a=*/false, a, /*neg_b=*/false, b,
            /*c_mod=*/(short)0, acc, /*reuse_a=*/false, /*reuse_b=*/false);
    }

    // C/D layout: VGPR v, lanes 0-15 -> M = v ; lanes 16-31 -> M = v+8
    const int mbase = m0 + ((lane < 16) ? 0 : 8);
    if (m0 + 16 <= N) {
        // full tile: unguarded stores (common case; N=50000 is 3125 full tiles)
#pragma unroll
        for (int v = 0; v < 8; ++v)
            H[(size_t)(mbase + v) * 128 + n0 + r] = acc[v];
    } else {
#pragma unroll
        for (int v = 0; v < 8; ++v) {
            int gr = mbase + v;
            if (gr < N) H[(size_t)gr * 128 + n0 + r] = acc[v];
        }
    }
}

// ---------------------------------------------------------- aggregation init
// agg[n,f] = dinv[n]^2 * h[n,f] + b[f]   (self-loop message + bias)
__global__ void gcn_selfloop_bias(const float* __restrict__ h,
                                  const float* __restrict__ dinv,
                                  const float* __restrict__ b,
                                  float* __restrict__ agg, int N) {
    int i = blockIdx.x * blockDim.x + threadIdx.x;
    if (i < N * 128) {
        int n = i >> 7, f = i & 127;
        float d = dinv[n];
        agg[i] = d * d * h[i] + b[f];
    }
}

// -------------------------------------------------------------- edge scatter
// One wave per edge: 32 lanes x float4 = 128 features.
// agg[col] += (dinv[row]*dinv[col]) * h[row]   via L2 f32 atomics.
__global__ __launch_bounds__(256)
void gcn_edge_scatter(const long long* __restrict__ row,
                      const long long* __restrict__ col,
                      const float* __restrict__ dinv,
                      const float* __restrict__ h,
                      float* __restrict__ agg, int E) {
    int e    = (int)((blockIdx.x * (size_t)blockDim.x + threadIdx.x) >> 5);
    int lane = threadIdx.x & 31;
    if (e >= E) return;

    int rs = (int)row[e];
    int cd = (int)col[e];
    float nrm = dinv[rs] * dinv[cd];

    const float4 m = ((const float4*)(h + (size_t)rs * 128))[lane];
    float* out = agg + (size_t)cd * 128 + lane * 4;
    __hip_atomic_fetch_add(out + 0, nrm * m.x, __ATOMIC_RELAXED, __HIP_MEMORY_SCOPE_AGENT);
    __hip_atomic_fetch_add(out + 1, nrm * m.y, __ATOMIC_RELAXED, __HIP_MEMORY_SCOPE_AGENT);
    __hip_atomic_fetch_add(out + 2, nrm * m.z, __ATOMIC_RELAXED, __HIP_MEMORY_SCOPE_AGENT);
    __hip_atomic_fetch_add(out + 3, nrm * m.w, __ATOMIC_RELAXED, __HIP_MEMORY_SCOPE_AGENT);
}

// ------------------------------------------------------------------- pooling
// gsum[f] += sum over strided rows of agg[n,f]
__global__ void gcn_pool_reduce(const float* __restrict__ agg,
                                float* __restrict__ gsum, int N) {
    int f = threadIdx.x;   // 0..127
    float s = 0.0f;
    for (int n = blockIdx.x; n < N; n += gridDim.x)
        s += agg[(size_t)n * 128 + f];
    __hip_atomic_fetch_add(&gsum[f], s, __ATOMIC_RELAXED,
                           __HIP_MEMORY_SCOPE_AGENT);
}

// out = sigmoid( (gsum/N) . Wl + bl )
__global__ void gcn_final(const float* __restrict__ gsum,
                          const float* __restrict__ Wl,
                          const float* __restrict__ bl,
                          float* __restrict__ out, int N) {
    if (threadIdx.x == 0 && blockIdx.x == 0) {
        float inv = 1.0f / (float)N;
        float acc = bl[0];
        for (int f = 0; f < 128; ++f) acc += gsum[f] * inv * Wl[f];
        out[0] = 1.0f / (1.0f + __expf(-acc));
    }
}

// =============================================================== launch glue
extern "C" void kernel_launch(void* const* d_in, const int* in_sizes, int n_in,
                              void* d_out, int out_size, void* d_ws, size_t ws_size,
                              hipStream_t stream) {
    const float*     x   = (const float*)d_in[0];
    const long long* ei  = (const long long*)d_in[1];   // int64 per reference
    // d_in[2] = pos, unused (matches reference)
    const float* W1 = (const float*)d_in[3];
    const float* b1 = (const float*)d_in[4];
    const float* W2 = (const float*)d_in[5];
    const float* b2 = (const float*)d_in[6];
    const float* W3 = (const float*)d_in[7];
    const float* b3 = (const float*)d_in[8];
    const float* Wl = (const float*)d_in[9];
    const float* bl = (const float*)d_in[10];
    float* out = (float*)d_out;

    const int N = in_sizes[0] / 128;
    const int E = in_sizes[1] / 2;
    const long long* row = ei;
    const long long* col = ei + E;

    // workspace layout (256B aligned): dinv[N] | h[N*128] | agg[N*128] | gsum[128]
    char* ws = (char*)d_ws;
    size_t off = 0;
    float* dinv = (float*)(ws + off); off += (((size_t)N * 4) + 255) & ~(size_t)255;
    float* h    = (float*)(ws + off); off += (((size_t)N * 128 * 4) + 255) & ~(size_t)255;
    float* agg  = (float*)(ws + off); off += (((size_t)N * 128 * 4) + 255) & ~(size_t)255;
    float* gsum = (float*)(ws + off);

    const int T = 256;
    const int gN    = (N + T - 1) / T;
    const int gE    = (E + T - 1) / T;
    const int gNF   = (N * 128 + T - 1) / T;
    const int gEdge = (int)(((size_t)E * 32 + T - 1) / T);
    const int gGemm = (N + 15) / 16;

    // --- normalization (shared across the 3 convs) ---
    gcn_zero_f32<<<gN, T, 0, stream>>>(dinv, N);
    gcn_degree<<<gE, T, 0, stream>>>(col, dinv, E);
    gcn_dinv<<<gN, T, 0, stream>>>(dinv, N);

    // --- layer 1 ---
    gcn_gemm_wmma<false><<<gGemm, T, 0, stream>>>(x, W1, h, N);
    gcn_selfloop_bias<<<gNF, T, 0, stream>>>(h, dinv, b1, agg, N);
    gcn_edge_scatter<<<gEdge, T, 0, stream>>>(row, col, dinv, h, agg, E);

    // --- layer 2 (ReLU of layer-1 output fused into fragment reads) ---
    gcn_gemm_wmma<true><<<gGemm, T, 0, stream>>>(agg, W2, h, N);
    gcn_selfloop_bias<<<gNF, T, 0, stream>>>(h, dinv, b2, agg, N);
    gcn_edge_scatter<<<gEdge, T, 0, stream>>>(row, col, dinv, h, agg, E);

    // --- layer 3 ---
    gcn_gemm_wmma<true><<<gGemm, T, 0, stream>>>(agg, W3, h, N);
    gcn_selfloop_bias<<<gNF, T, 0, stream>>>(h, dinv, b3, agg, N);
    gcn_edge_scatter<<<gEdge, T, 0, stream>>>(row, col, dinv, h, agg, E);

    // --- global mean pool + linear + sigmoid ---
    gcn_zero_f32<<<1, 128, 0, stream>>>(gsum, 128);
    gcn_pool_reduce<<<512, 128, 0, stream>>>(agg, gsum, N);
    gcn_final<<<1, 32, 0, stream>>>(gsum, Wl, bl, out, N);
}